// Net_1254130451160
// MI455X (gfx1250) — compile-verified
//
#include <hip/hip_runtime.h>

typedef __attribute__((ext_vector_type(2))) float v2f;
typedef __attribute__((ext_vector_type(8))) float v8f;

#define NNODES 100000
#define NEDGES 1600000

// ---------------------------------------------------------------------------
// Degree: deg[t] += 1 per edge, then invert in place: deg[i] = 1/max(deg,1)
// ---------------------------------------------------------------------------
__global__ void deg_kernel(const long long* __restrict__ tgt, float* __restrict__ deg, int E) {
    int i = blockIdx.x * blockDim.x + threadIdx.x;
    if (i < E) atomicAdd(&deg[(int)tgt[i]], 1.0f);
}

__global__ void invdeg_kernel(float* __restrict__ deg, int n) {
    int i = blockIdx.x * blockDim.x + threadIdx.x;
    if (i < n) deg[i] = 1.0f / fmaxf(deg[i], 1.0f);
}

// ---------------------------------------------------------------------------
// Layer-1 scatter: agg[t, f] += x[s, f] for f in 0..8
// ---------------------------------------------------------------------------
__global__ void scatter9(const long long* __restrict__ src, const long long* __restrict__ tgt,
                         const float* __restrict__ x, float* __restrict__ agg, int E) {
    int i = blockIdx.x * blockDim.x + threadIdx.x;
    if (i >= E * 9) return;
    int e = i / 9;
    int f = i - e * 9;
    int s = (int)src[e];
    int t = (int)tgt[e];
    atomicAdd(&agg[t * 9 + f], x[s * 9 + f]);
}

// ---------------------------------------------------------------------------
// 128-wide scatter: one thread moves 4 contiguous features of one edge.
// float4 global load + 4 non-returning atomic f32 adds (agg lives in L2).
// ---------------------------------------------------------------------------
__global__ void scatter128(const long long* __restrict__ src, const long long* __restrict__ tgt,
                           const float* __restrict__ h, float* __restrict__ agg, int E) {
    long i = (long)blockIdx.x * blockDim.x + threadIdx.x;
    if (i >= (long)E * 32) return;
    int e = (int)(i >> 5);
    int c = ((int)i & 31) << 2;
    int s = (int)src[e];
    int t = (int)tgt[e];
    const float4 v = *(const float4*)(h + (long)s * 128 + c);
    float* a = agg + (long)t * 128 + c;
    atomicAdd(a + 0, v.x);
    atomicAdd(a + 1, v.y);
    atomicAdd(a + 2, v.z);
    atomicAdd(a + 3, v.w);
}

// ---------------------------------------------------------------------------
// Layer 1 (K=9, too small for WMMA): one block per node, one thread per out
// feature. out = relu(aggm @ Wl + x @ Wr + b)
// ---------------------------------------------------------------------------
__global__ void sage_layer1(const float* __restrict__ x, const float* __restrict__ agg,
                            const float* __restrict__ invdeg,
                            const float* __restrict__ Wl, const float* __restrict__ Wr,
                            const float* __restrict__ bias, float* __restrict__ out) {
    const int i = blockIdx.x;     // node
    const int j = threadIdx.x;    // out feature (128 threads)
    __shared__ float xs[9];
    __shared__ float as[9];
    if (j < 9) {
        xs[j] = x[i * 9 + j];
        as[j] = agg[i * 9 + j] * invdeg[i];
    }
    __syncthreads();
    float s = bias[j];
#pragma unroll
    for (int k = 0; k < 9; ++k) {
        s += as[k] * Wl[k * 128 + j];
        s += xs[k] * Wr[k * 128 + j];
    }
    out[(long)i * 128 + j] = fmaxf(s, 0.0f);
}

// ---------------------------------------------------------------------------
// Fused dual-GEMM with full-fp32 WMMA (V_WMMA_F32_16X16X4_F32).
//   out = relu?( A1*scale1 @ W1 [+ A2 @ W2] + bias )
// Compile-time K / NOUT -> fully unrolled K-loop, immediate-offset loads.
// Two independent accumulator chains so loads/WMMAs pipeline.
// One wave -> one 16x16 tile; block = (NOUT/16) waves over 16 rows.
// A-frag layout (16x4 f32): lanes 0-15 K={k0,k0+1}, lanes 16-31 K={k0+2,k0+3}.
// C/D layout: vgpr r -> M=r (lanes 0-15) / M=r+8 (lanes 16-31), N = lane&15.
// ---------------------------------------------------------------------------
template <int K, int NOUT, bool DUAL, bool RELU, bool SCALE>
__global__ void gemm_wmma(const float* __restrict__ A1, const float* __restrict__ scale1,
                          const float* __restrict__ W1,
                          const float* __restrict__ A2, const float* __restrict__ W2,
                          const float* __restrict__ bias, float* __restrict__ out) {
    __shared__ float As1[16][K + 4];
    __shared__ float As2[DUAL ? 16 : 1][K + 4];

    const int tid = threadIdx.x;
    const int m0  = blockIdx.x * 16;

    // Cooperative stage of the 16-row A tiles (scale A1 by invdeg while loading).
    for (int i = tid; i < 16 * K; i += blockDim.x) {
        const int r = i / K;
        const int c = i - r * K;
        float s = SCALE ? scale1[m0 + r] : 1.0f;
        As1[r][c] = A1[(long)(m0 + r) * K + c] * s;
        if (DUAL) As2[r][c] = A2[(long)(m0 + r) * K + c];
    }
    __syncthreads();

    const int lane = tid & 31;
    const int wave = tid >> 5;
    const int n0   = wave * 16;
    const int l15  = lane & 15;
    const int koff = (lane >= 16) ? 2 : 0;

    // Two accumulator chains; bias (column-only) seeds chain 0.
    const float bv = bias[n0 + l15];
    v8f acc0, acc1;
#pragma unroll
    for (int r = 0; r < 8; ++r) { acc0[r] = bv; acc1[r] = 0.0f; }

    // Fixed per-wave B base pointers; all k offsets are compile-time.
    const float* wp1 = W1 + (long)koff * NOUT + n0 + l15;
    const float* wp2 = DUAL ? (W2 + (long)koff * NOUT + n0 + l15) : nullptr;

#pragma unroll
    for (int k0 = 0; k0 < K; k0 += 8) {
        v2f a, b;
        // chunk k0 -> chain 0
        a[0] = As1[l15][k0 + koff];
        a[1] = As1[l15][k0 + koff + 1];
        b[0] = wp1[k0 * NOUT];
        b[1] = wp1[(k0 + 1) * NOUT];
        acc0 = __builtin_amdgcn_wmma_f32_16x16x4_f32(false, a, false, b, (short)0, acc0,
                                                     false, false);
        // chunk k0+4 -> chain 1
        a[0] = As1[l15][k0 + 4 + koff];
        a[1] = As1[l15][k0 + 4 + koff + 1];
        b[0] = wp1[(k0 + 4) * NOUT];
        b[1] = wp1[(k0 + 5) * NOUT];
        acc1 = __builtin_amdgcn_wmma_f32_16x16x4_f32(false, a, false, b, (short)0, acc1,
                                                     false, false);
    }

    if (DUAL) {
#pragma unroll
        for (int k0 = 0; k0 < K; k0 += 8) {
            v2f a, b;
            a[0] = As2[l15][k0 + koff];
            a[1] = As2[l15][k0 + koff + 1];
            b[0] = wp2[k0 * NOUT];
            b[1] = wp2[(k0 + 1) * NOUT];
            acc0 = __builtin_amdgcn_wmma_f32_16x16x4_f32(false, a, false, b, (short)0, acc0,
                                                         false, false);
            a[0] = As2[l15][k0 + 4 + koff];
            a[1] = As2[l15][k0 + 4 + koff + 1];
            b[0] = wp2[(k0 + 4) * NOUT];
            b[1] = wp2[(k0 + 5) * NOUT];
            acc1 = __builtin_amdgcn_wmma_f32_16x16x4_f32(false, a, false, b, (short)0, acc1,
                                                         false, false);
        }
    }

    const int rbase = (lane >= 16) ? 8 : 0;
#pragma unroll
    for (int r = 0; r < 8; ++r) {
        float v = acc0[r] + acc1[r];
        if (RELU) v = fmaxf(v, 0.0f);
        out[(long)(m0 + r + rbase) * NOUT + n0 + l15] = v;
    }
}

// ---------------------------------------------------------------------------
// Head: out = log_softmax(t2 @ Wl3 + bl3), K=64, 3 classes. One thread/node.
// ---------------------------------------------------------------------------
__global__ void head_kernel(const float* __restrict__ t2, const float* __restrict__ W,
                            const float* __restrict__ b, float* __restrict__ out, int M) {
    int i = blockIdx.x * blockDim.x + threadIdx.x;
    if (i >= M) return;
    float a0 = b[0], a1 = b[1], a2 = b[2];
    const float* row = t2 + (long)i * 64;
#pragma unroll 4
    for (int k = 0; k < 64; ++k) {
        float v = row[k];
        a0 += v * W[k * 3 + 0];
        a1 += v * W[k * 3 + 1];
        a2 += v * W[k * 3 + 2];
    }
    float m   = fmaxf(a0, fmaxf(a1, a2));
    float lse = m + logf(expf(a0 - m) + expf(a1 - m) + expf(a2 - m));
    out[(long)i * 3 + 0] = a0 - lse;
    out[(long)i * 3 + 1] = a1 - lse;
    out[(long)i * 3 + 2] = a2 - lse;
}

// ---------------------------------------------------------------------------
extern "C" void kernel_launch(void* const* d_in, const int* in_sizes, int n_in,
                              void* d_out, int out_size, void* d_ws, size_t ws_size,
                              hipStream_t stream) {
    const float*     x    = (const float*)d_in[0];
    const long long* ei   = (const long long*)d_in[1];  // int64 per reference
    const long long* src  = ei;
    const long long* tgt  = ei + NEDGES;
    const float* W1_l = (const float*)d_in[2];
    const float* b1   = (const float*)d_in[3];
    const float* W1_r = (const float*)d_in[4];
    const float* W2_l = (const float*)d_in[5];
    const float* b2   = (const float*)d_in[6];
    const float* W2_r = (const float*)d_in[7];
    const float* W3_l = (const float*)d_in[8];
    const float* b3   = (const float*)d_in[9];
    const float* W3_r = (const float*)d_in[10];
    const float* Wl1  = (const float*)d_in[11];
    const float* bl1  = (const float*)d_in[12];
    const float* Wl2  = (const float*)d_in[13];
    const float* bl2  = (const float*)d_in[14];
    const float* Wl3  = (const float*)d_in[15];
    const float* bl3  = (const float*)d_in[16];
    float* out = (float*)d_out;

    // Workspace layout (floats): deg[N] | agg[N*128] | hA[N*128] | hB[N*128]
    float* deg = (float*)d_ws;
    float* agg = deg + 102400;                 // padded past N
    float* hA  = agg + (long)NNODES * 128;
    float* hB  = hA  + (long)NNODES * 128;

    const int TB = 256;

    // Degree -> invdeg (shared by all three SAGE layers)
    hipMemsetAsync(deg, 0, (size_t)NNODES * sizeof(float), stream);
    deg_kernel<<<(NEDGES + TB - 1) / TB, TB, 0, stream>>>(tgt, deg, NEDGES);
    invdeg_kernel<<<(NNODES + TB - 1) / TB, TB, 0, stream>>>(deg, NNODES);

    // ---- Layer 1 (F=9 -> 128) ----
    hipMemsetAsync(agg, 0, (size_t)NNODES * 9 * sizeof(float), stream);
    scatter9<<<(NEDGES * 9 + TB - 1) / TB, TB, 0, stream>>>(src, tgt, x, agg, NEDGES);
    sage_layer1<<<NNODES, 128, 0, stream>>>(x, agg, deg, W1_l, W1_r, b1, hA);

    const int gemm_blocks = NNODES / 16;                       // 6250, exact
    const int sc_blocks   = (int)(((long)NEDGES * 32 + TB - 1) / TB);

    // ---- Layer 2 (128 -> 128) ----
    hipMemsetAsync(agg, 0, (size_t)NNODES * 128 * sizeof(float), stream);
    scatter128<<<sc_blocks, TB, 0, stream>>>(src, tgt, hA, agg, NEDGES);
    gemm_wmma<128, 128, true, true, true>
        <<<gemm_blocks, 256, 0, stream>>>(agg, deg, W2_l, hA, W2_r, b2, hB);

    // ---- Layer 3 (128 -> 128) ----
    hipMemsetAsync(agg, 0, (size_t)NNODES * 128 * sizeof(float), stream);
    scatter128<<<sc_blocks, TB, 0, stream>>>(src, tgt, hB, agg, NEDGES);
    gemm_wmma<128, 128, true, true, true>
        <<<gemm_blocks, 256, 0, stream>>>(agg, deg, W3_l, hB, W3_r, b3, hA);

    // ---- MLP head ----
    gemm_wmma<128, 128, false, true, false>
        <<<gemm_blocks, 256, 0, stream>>>(hA, nullptr, Wl1, nullptr, nullptr, bl1, hB);
    gemm_wmma<128, 64, false, true, false>
        <<<gemm_blocks, 128, 0, stream>>>(hB, nullptr, Wl2, nullptr, nullptr, bl2, agg);
    head_kernel<<<(NNODES + TB - 1) / TB, TB, 0, stream>>>(agg, Wl3, bl3, out, NNODES);
}